// PerforantPathSymphonyBridge_25262997636013
// MI455X (gfx1250) — compile-verified
//
#include <hip/hip_runtime.h>
#include <math.h>

typedef __attribute__((ext_vector_type(2))) float v2f;
typedef __attribute__((ext_vector_type(8))) float v8f;

namespace {

constexpr int kNCols = 6;
constexpr int kB = 16384;
constexpr int kD = 2048;
constexpr int kC = 64;
constexpr int kRowsPerTile = 16;   // one 16x16 WMMA tile of batch rows
constexpr int kTilesPerBlock = 4;  // 8 waves = 4 tiles x 2 split-K waves

__device__ __forceinline__ v8f wmma4(v2f a, v2f b, v8f c) {
  // V_WMMA_F32_16X16X4_F32 : D = A(16x4) * B(4x16) + C(16x16), all f32
  return __builtin_amdgcn_wmma_f32_16x16x4_f32(false, a, false, b, (short)0, c,
                                               false, false);
}

__device__ __forceinline__ v2f ntload2(const float* p) {
  return __builtin_nontemporal_load(reinterpret_cast<const v2f*>(p));
}

__global__ __launch_bounds__(256, 1) void perforant_fused(
    const float* __restrict__ colout,  // [6, 16384, 2048]
    const float* __restrict__ prime,   // [16384, 2048]
    const float* __restrict__ U,       // [6, 2048, 3]
    const float* __restrict__ gains,   // [6, 3]
    const float* __restrict__ V,       // [6, 3, 64]
    const float* __restrict__ Wq,      // [64, 2048]
    float* __restrict__ out)           // coords [16384,64] ++ weights [16384,6]
{
  // Split-K partial accumulators (v8f per lane) + subspace tiles for epilogue.
  __shared__ __attribute__((aligned(32))) float partial[kTilesPerBlock][10][32][8];
  __shared__ __attribute__((aligned(16))) float subLds[kTilesPerBlock][kNCols][16][4];

  const int tid   = (int)threadIdx.x;
  const int wave  = tid >> 5;
  const int lane  = tid & 31;
  const int l     = lane & 15;  // row-in-tile / B-column lane
  const int h     = lane >> 4;  // K-half selector (ISA A/B layout)
  const int pair  = wave >> 1;  // row-tile within block
  const int kHalf = wave & 1;   // split-K id
  const int rowBase = ((int)blockIdx.x * kTilesPerBlock + pair) * kRowsPerTile;
  const int row     = rowBase + l;
  const int rClamp  = (l < 2) ? l : 2;  // valid address for B cols >= R

  // Per-lane K base: split-K half + the 2h K-offset demanded by the WMMA layout.
  const int kBase = kHalf * (kD / 2) + 2 * h;

  const float* pA = prime + (size_t)row * kD + kBase;
  const float* pW[4];
#pragma unroll
  for (int cb = 0; cb < 4; ++cb)
    pW[cb] = Wq + (size_t)(cb * 16 + l) * kD + kBase;
  const float* pC[kNCols];
  const float* pU[kNCols];
#pragma unroll
  for (int n = 0; n < kNCols; ++n) {
    pC[n] = colout + ((size_t)n * kB + row) * kD + kBase;
    pU[n] = U + (size_t)n * kD * 3 + (size_t)kBase * 3 + rClamp;
  }

  const v8f zf = {0.f, 0.f, 0.f, 0.f, 0.f, 0.f, 0.f, 0.f};
  v8f qa[4];
  v8f sa[kNCols];
#pragma unroll
  for (int cb = 0; cb < 4; ++cb) qa[cb] = zf;
#pragma unroll
  for (int n = 0; n < kNCols; ++n) sa[n] = zf;

  // ---- main streaming loop: this wave reduces kD/2 of the contraction ----
#pragma unroll 2
  for (int k = 0; k < kD / 2; k += 4) {
    v2f a = ntload2(pA + k);  // prime rows, NT (stream once)
    v2f wv[4];
#pragma unroll
    for (int cb = 0; cb < 4; ++cb)  // Wq: small, keep cached (RT)
      wv[cb] = *reinterpret_cast<const v2f*>(pW[cb] + k);
    v2f cv[kNCols];
#pragma unroll
    for (int n = 0; n < kNCols; ++n)  // column_outputs, NT (stream once)
      cv[n] = ntload2(pC[n] + k);
    v2f uv[kNCols];
#pragma unroll
    for (int n = 0; n < kNCols; ++n) {  // U: [k][r] pairs, stride R=3 floats
      uv[n].x = pU[n][3 * k];
      uv[n].y = pU[n][3 * k + 3];
    }
#pragma unroll
    for (int cb = 0; cb < 4; ++cb) qa[cb] = wmma4(a, wv[cb], qa[cb]);
#pragma unroll
    for (int n = 0; n < kNCols; ++n) sa[n] = wmma4(cv[n], uv[n], sa[n]);
  }

  // ---- combine split-K partials through LDS ----
  if (kHalf == 1) {
#pragma unroll
    for (int cb = 0; cb < 4; ++cb)
      *reinterpret_cast<v8f*>(&partial[pair][cb][lane][0]) = qa[cb];
#pragma unroll
    for (int n = 0; n < kNCols; ++n)
      *reinterpret_cast<v8f*>(&partial[pair][4 + n][lane][0]) = sa[n];
  }
  __syncthreads();
  if (kHalf == 1) return;  // helper wave done; tile wave finishes alone

#pragma unroll
  for (int cb = 0; cb < 4; ++cb)
    qa[cb] += *reinterpret_cast<const v8f*>(&partial[pair][cb][lane][0]);
#pragma unroll
  for (int n = 0; n < kNCols; ++n)
    sa[n] += *reinterpret_cast<const v8f*>(&partial[pair][4 + n][lane][0]);

  // ---- subspace accs -> LDS in A-matrix-friendly [row][r(pad4)] form ----
  // acc layout: element j of lane (l,h) is (row = j + 8h, r = l); r==3 pad = 0
#pragma unroll
  for (int n = 0; n < kNCols; ++n) {
#pragma unroll
    for (int j = 0; j < 8; ++j) {
      if (l < 4) subLds[pair][n][j + 8 * h][l] = (l < 3) ? sa[n][j] : 0.0f;
    }
  }

  // gains folded into V:  stacked = subspace @ (diag(g) V)
  float g[kNCols][3];
#pragma unroll
  for (int n = 0; n < kNCols; ++n)
#pragma unroll
    for (int r = 0; r < 3; ++r) g[n][r] = gains[n * 3 + r];

  // A operands for the stacked WMMAs: lane holds subspace[row=l][K = 2h, 2h+1]
  v2f av[kNCols];
#pragma unroll
  for (int n = 0; n < kNCols; ++n)
    av[n] = *reinterpret_cast<const v2f*>(&subLds[pair][n][l][2 * h]);

  // B builder: B[K=2h..2h+1][col=c] = g*V[n][r][c]; K=3 row is zero (A pad also 0)
  auto makeB = [&](int n, int cb) -> v2f {
    v2f b;
    b.x = g[n][2 * h] * V[(n * 3 + 2 * h) * kC + cb * 16 + l];
    b.y = (h == 0) ? g[n][1] * V[(n * 3 + 1) * kC + cb * 16 + l] : 0.0f;
    return b;
  };

  // ---- scores: s[b][n] = (1/8) * sum_c q[b][c] * stacked[b][n][c] ----
  float w[kNCols][8];
#pragma unroll
  for (int n = 0; n < kNCols; ++n) {
    v8f st[4];
#pragma unroll
    for (int cb = 0; cb < 4; ++cb) st[cb] = wmma4(av[n], makeB(n, cb), zf);
#pragma unroll
    for (int j = 0; j < 8; ++j) {
      float p = qa[0][j] * st[0][j] + qa[1][j] * st[1][j] +
                qa[2][j] * st[2][j] + qa[3][j] * st[3][j];
      // reduce over the 16 c-lanes of this half-wave (rows differ per half)
      p += __shfl_xor(p, 1);
      p += __shfl_xor(p, 2);
      p += __shfl_xor(p, 4);
      p += __shfl_xor(p, 8);
      w[n][j] = p * 0.125f;  // 1/sqrt(C)
    }
  }

  // ---- softmax over n (replicated across the 16 lanes of each half) ----
#pragma unroll
  for (int j = 0; j < 8; ++j) {
    float m = w[0][j];
#pragma unroll
    for (int n = 1; n < kNCols; ++n) m = fmaxf(m, w[n][j]);
    float s = 0.f;
#pragma unroll
    for (int n = 0; n < kNCols; ++n) {
      float e = __expf(w[n][j] - m);
      w[n][j] = e;
      s += e;
    }
    float inv = 1.0f / s;
#pragma unroll
    for (int n = 0; n < kNCols; ++n) w[n][j] *= inv;
  }

  // ---- coords: recompute stacked per n, weight and accumulate ----
  v8f ca[4];
#pragma unroll
  for (int cb = 0; cb < 4; ++cb) ca[cb] = zf;
#pragma unroll
  for (int n = 0; n < kNCols; ++n) {
    v8f st[4];
#pragma unroll
    for (int cb = 0; cb < 4; ++cb) st[cb] = wmma4(av[n], makeB(n, cb), zf);
#pragma unroll
    for (int cb = 0; cb < 4; ++cb)
#pragma unroll
      for (int j = 0; j < 8; ++j) ca[cb][j] = fmaf(w[n][j], st[cb][j], ca[cb][j]);
  }

  // ---- stores: coords [B,64] then attn_weights [B,6] ----
#pragma unroll
  for (int cb = 0; cb < 4; ++cb)
#pragma unroll
    for (int j = 0; j < 8; ++j)
      out[(size_t)(rowBase + j + 8 * h) * kC + cb * 16 + l] = ca[cb][j];

  float* wout = out + (size_t)kB * kC;
#pragma unroll
  for (int j = 0; j < 8; ++j) {
    float wv = (l == 0) ? w[0][j]
             : (l == 1) ? w[1][j]
             : (l == 2) ? w[2][j]
             : (l == 3) ? w[3][j]
             : (l == 4) ? w[4][j]
                        : w[5][j];
    if (l < 6) wout[(size_t)(rowBase + j + 8 * h) * kNCols + l] = wv;
  }
}

}  // namespace

extern "C" void kernel_launch(void* const* d_in, const int* in_sizes, int n_in,
                              void* d_out, int out_size, void* d_ws,
                              size_t ws_size, hipStream_t stream) {
  (void)in_sizes; (void)n_in; (void)out_size; (void)d_ws; (void)ws_size;
  const float* colout = (const float*)d_in[0];
  const float* prime  = (const float*)d_in[1];
  const float* U      = (const float*)d_in[2];
  const float* gains  = (const float*)d_in[3];
  const float* V      = (const float*)d_in[4];
  const float* Wq     = (const float*)d_in[5];
  float* out = (float*)d_out;

  dim3 grid(kB / (kTilesPerBlock * kRowsPerTile));  // 256 blocks
  dim3 block(256);                                  // 8 waves (wave32)
  hipLaunchKernelGGL(perforant_fused, grid, block, 0, stream, colout, prime, U,
                     gains, V, Wq, out);
}